// AffinityDisplacementLoss_5901285064958
// MI455X (gfx1250) — compile-verified
//
#include <hip/hip_runtime.h>
#include <hip/hip_bf16.h>

typedef __attribute__((ext_vector_type(2))) float v2f;
typedef __attribute__((ext_vector_type(8))) float v8f;

#define NB   512   // stage-1 blocks (multiple of 8 for the WMMA reduce)
#define TPB  256
#define MAXL 16    // cached path length (Lmax ~ 11 for RADIUS=5)

// ---------------------------------------------------------------------------
// Stage 1: fused gather + loss + partial reduction.
// Thread owns (d,p) pairs (grid-stride); batch loop innermost so the
// batch-invariant index gathers (path_idx/src/dst/disp_target) are loaded once.
// ---------------------------------------------------------------------------
__global__ __launch_bounds__(TPB) void loss_stage1(
    const float* __restrict__ edge, const float* __restrict__ dp,
    const float* __restrict__ bg,  const float* __restrict__ fg,
    const float* __restrict__ ng,  const float* __restrict__ dispt,
    const int* __restrict__ path_idx, const int* __restrict__ src_idx,
    const int* __restrict__ dst_idx,
    int B, int HW, int n_dst, int n_pos, int Lmax,
    float* __restrict__ ws)
{
    const float eps = 1e-5f;
    const int   total = n_dst * n_pos;
    const long  bstride = (long)n_dst * n_pos;

    float acc[8];
#pragma unroll
    for (int j = 0; j < 8; ++j) acc[j] = 0.0f;

    for (int t = blockIdx.x * TPB + threadIdx.x; t < total; t += NB * TPB) {
        const int d = t / n_pos;
        const int p = t - d * n_pos;

        // Batch-invariant gathers: load once, reuse for all B batches.
        int pidx[MAXL];
        const int Lc = (Lmax < MAXL) ? Lmax : MAXL;
#pragma unroll
        for (int l = 0; l < MAXL; ++l)
            if (l < Lc) pidx[l] = path_idx[(long)(d * Lmax + l) * n_pos + p];
        const int   s  = src_idx[p];
        const int   di = dst_idx[(long)d * n_pos + p];
        const float t0 = dispt[d];            // disp_target (1,2,n_dst,1)
        const float t1 = dispt[n_dst + d];

        for (int b = 0; b < B; ++b) {
            const long  lab = (long)b * bstride + t;
            if (b + 1 < B) {                  // CDNA5: global_prefetch_b8
                __builtin_prefetch(&bg[lab + bstride], 0, 1);
                __builtin_prefetch(&fg[lab + bstride], 0, 1);
                __builtin_prefetch(&ng[lab + bstride], 0, 1);
            }
            const float* eb = edge + (long)b * HW;
            // max(sigmoid(x)) == sigmoid(max(x)): fold transcendentals.
            float m = eb[pidx[0]];
            for (int l = 1; l < Lmax; ++l) {
                float v = (l < MAXL) ? eb[pidx[l]]
                                     : eb[path_idx[(long)(d * Lmax + l) * n_pos + p]];
                m = fmaxf(m, v);
            }
            const float aff  = 1.0f / (1.0f + __expf(m));      // 1 - sigmoid(m)
            const float posl = -__logf(aff + eps);
            const float negl = -__logf(1.0f + eps - aff);

            const float vbg = bg[lab], vfg = fg[lab], vng = ng[lab];

            const float* dpb = dp + (long)b * 2 * HW;
            const float pd0 = dpb[s] - dpb[di];
            const float pd1 = dpb[HW + s] - dpb[HW + di];
            const float fgdp = fabsf(pd0 - t0) + fabsf(pd1 - t1);
            const float bgdp = fabsf(pd0) + fabsf(pd1);

            acc[0] += vbg * posl;  acc[1] += vbg;
            acc[2] += vfg * posl;  acc[3] += vfg;
            acc[4] += vng * negl;  acc[5] += vng;
            acc[6] += vfg * fgdp;  acc[7] += vbg * bgdp;
        }
    }

    // wave32 shuffle reduction, then cross-wave via LDS (deterministic order).
#pragma unroll
    for (int off = 16; off > 0; off >>= 1)
#pragma unroll
        for (int j = 0; j < 8; ++j)
            acc[j] += __shfl_down(acc[j], off, 32);

    __shared__ float red[(TPB / 32) * 8];
    const int lane = threadIdx.x & 31, wid = threadIdx.x >> 5;
    if (lane == 0)
#pragma unroll
        for (int j = 0; j < 8; ++j) red[wid * 8 + j] = acc[j];
    __syncthreads();
    if (threadIdx.x < 8) {
        float s = 0.0f;
#pragma unroll
        for (int w = 0; w < TPB / 32; ++w) s += red[w * 8 + threadIdx.x];
        ws[blockIdx.x * 8 + threadIdx.x] = s;
    }
}

// ---------------------------------------------------------------------------
// Stage 2: one wave32. Reduce NB x 8 partials with V_WMMA_F32_16X16X4_F32:
// A(16x4) holds 64 partials/iter (row m = acc j=m&7, group g=m>>3; K picks the
// block), B = all-ones (layout-independent), C accumulates across iterations.
// total[j] = D[j][0] + D[j+8][0]  (lane0 / lane16, any column).
// ---------------------------------------------------------------------------
__global__ __launch_bounds__(32) void loss_stage2(
    const float* __restrict__ ws, int nblk, float* __restrict__ out)
{
    const int L  = threadIdx.x;          // 0..31, EXEC all ones
    const int j  = L & 7;
    const int g  = (L >> 3) & 1;
    const int k0 = (L >> 4) * 2;         // this lane's K pair

    v2f ones; ones.x = 1.0f; ones.y = 1.0f;
    v8f c = {};
    const int iters = nblk >> 3;         // 8 partial-vectors per WMMA
    for (int it = 0; it < iters; ++it) {
        const int base = it * 8;
        v2f a;
        a.x = ws[(base + k0 * 2 + g) * 8 + j];       // A[m][k0]
        a.y = ws[(base + k0 * 2 + 2 + g) * 8 + j];   // A[m][k0+1]
        c = __builtin_amdgcn_wmma_f32_16x16x4_f32(
                false, a, false, ones, (short)0, c, false, false);
    }

    __shared__ float tot[16];
    if ((L & 15) == 0) {
        const int h = L >> 4;
#pragma unroll
        for (int r = 0; r < 8; ++r) tot[h * 8 + r] = c[r];
    }
    __syncthreads();
    if (L == 0) {
        float S[8];
#pragma unroll
        for (int r = 0; r < 8; ++r) S[r] = tot[r] + tot[8 + r];
        const float eps = 1e-5f;
        const float bg_pos = S[0] / (S[1] + eps);
        const float fg_pos = S[2] / (S[3] + eps);
        const float pos    = 0.5f * (bg_pos + fg_pos);
        const float neg    = S[4] / (S[5] + eps);
        const float dp_fg  = S[6] / (2.0f * S[3] + eps);
        const float dp_bg  = S[7] / (2.0f * S[1] + eps);
        out[0] = 0.5f * (pos + neg) + 0.5f * (dp_fg + dp_bg);
    }
}

extern "C" void kernel_launch(void* const* d_in, const int* in_sizes, int n_in,
                              void* d_out, int out_size, void* d_ws, size_t ws_size,
                              hipStream_t stream) {
    const float* edge  = (const float*)d_in[0];
    const float* dp    = (const float*)d_in[1];
    const float* bg    = (const float*)d_in[2];
    const float* fg    = (const float*)d_in[3];
    const float* ng    = (const float*)d_in[4];
    const float* dispt = (const float*)d_in[5];
    const int*   pidx  = (const int*)d_in[6];
    const int*   sidx  = (const int*)d_in[7];
    const int*   didx  = (const int*)d_in[8];

    const int n_pos = in_sizes[7];
    const int n_dst = in_sizes[8] / n_pos;
    const int Lmax  = in_sizes[6] / (n_dst * n_pos);
    const int B     = in_sizes[2] / (n_dst * n_pos);
    const int HW    = in_sizes[0] / B;

    float* ws = (float*)d_ws;            // NB*8 floats = 16 KB of scratch

    loss_stage1<<<NB, TPB, 0, stream>>>(edge, dp, bg, fg, ng, dispt,
                                        pidx, sidx, didx,
                                        B, HW, n_dst, n_pos, Lmax, ws);
    loss_stage2<<<1, 32, 0, stream>>>(ws, NB, (float*)d_out);
}